// TreeEncoderNet_44951127720315
// MI455X (gfx1250) — compile-verified
//
#include <hip/hip_runtime.h>
#include <hip/hip_bf16.h>

typedef __attribute__((ext_vector_type(2))) float v2f;
typedef __attribute__((ext_vector_type(8))) float v8f;

#define HID        256
#define NNODES     65535
#define KX         256      // embedding width
#define KH         512      // concat child h width
#define KB         768      // fused weight row length (KX + KH)
#define NCOLS_FULL 1280     // iou(768) + f0(256) + f1(256)

__device__ __forceinline__ float sigm(float x) { return 1.0f / (1.0f + expf(-x)); }

// Build Wbig (1280 x 768 row-major) and fused bias (1280) in workspace.
// Row r < 768:        [ W_iou_x[r][0:256] | W_iou_h[r][0:512] ]
// Row 768+t  (t<256): [ W_f_x[t][0:256]   | W_f_h[0][t][0:512] ]
// Row 1024+t (t<256): [ W_f_x[t][0:256]   | W_f_h[1][t][0:512] ]
__global__ void prep_weights(const float* __restrict__ Wioux, const float* __restrict__ bioux,
                             const float* __restrict__ Wfx,   const float* __restrict__ bfx,
                             const float* __restrict__ Wiouh, const float* __restrict__ Wfh,
                             float* __restrict__ Wbig, float* __restrict__ bias)
{
    int idx = blockIdx.x * blockDim.x + threadIdx.x;
    if (idx < NCOLS_FULL)
        bias[idx] = (idx < 768) ? bioux[idx] : bfx[(idx - 768) & 255];
    if (idx >= NCOLS_FULL * KB) return;
    int r = idx / KB, c = idx % KB;
    float v;
    if (r < 768) {
        v = (c < KX) ? Wioux[r * KX + c] : Wiouh[(size_t)r * KH + (c - KX)];
    } else {
        int k = (r - 768) >> 8;    // child select: 0 or 1
        int t = (r - 768) & 255;
        v = (c < KX) ? Wfx[t * KX + c]
                     : Wfh[((size_t)k * 256 + t) * KH + (c - KX)];
    }
    Wbig[idx] = v;
}

// Fused level kernel, LDS-free:
//  - block = 32 nodes (2 row-tiles); wave wv owns hidden dims t in [32*wv, 32*wv+32)
//  - wave computes column stripes 256*g + 32*wv (+16*s), g=0..NG-1, so all gate
//    components for dim t live in the SAME lane of its WMMA accumulators
//  - two row-tiles share every B fragment (halves L2 weight streaming)
//  - gates evaluated straight from accumulator registers; h -> d_out, c -> cbuf
template<int NG, int HASCH>   // NG=3 leaf (iou), NG=5 internal (iou+f0+f1)
__global__ void __launch_bounds__(256)
fused_level(const int* __restrict__ tokens, const float* __restrict__ emb,
            const float* __restrict__ Wbig, const float* __restrict__ bias,
            float* __restrict__ hbuf, float* __restrict__ cbuf,
            int levelOff, int n)
{
    const int lane = threadIdx.x & 31;
    const int half = lane >> 4;          // 0: K pair {k,k+1}, 1: {k+2,k+3}
    const int l    = lane & 15;
    const int wv   = threadIdx.x >> 5;   // column group 0..7 (dims 32*wv..32*wv+31)
    const int row0 = blockIdx.x << 5;    // 32 rows per block
    const int ko   = 2 * half;

    int r0 = row0 + l;      if (r0 >= n) r0 = n - 1;   // branchless clamps
    int r1 = row0 + 16 + l; if (r1 >= n) r1 = n - 1;
    const int m0 = levelOff + r0;
    const int m1 = levelOff + r1;

    const float* __restrict__ ax0 = emb  + (size_t)tokens[m0] * KX;
    const float* __restrict__ ax1 = emb  + (size_t)tokens[m1] * KX;
    const float* __restrict__ bw  = Wbig + (size_t)(32 * wv + l) * KB;  // tile offsets fold to imm24

    v8f acc[2][NG][2];
#pragma unroll
    for (int rt = 0; rt < 2; ++rt)
#pragma unroll
        for (int g = 0; g < NG; ++g)
#pragma unroll
            for (int s = 0; s < 2; ++s) acc[rt][g][s] = (v8f){};

    // ---- x part: K = 0..255 ----
    for (int k = 0; k < KX; k += 4) {
        v2f a0 = *(const v2f*)(ax0 + k + ko);
        v2f a1 = *(const v2f*)(ax1 + k + ko);
#pragma unroll
        for (int g = 0; g < NG; ++g)
#pragma unroll
            for (int s = 0; s < 2; ++s) {
                v2f wgt = *(const v2f*)(bw + (size_t)(256 * g + 16 * s) * KB + k + ko);
                acc[0][g][s] = __builtin_amdgcn_wmma_f32_16x16x4_f32(false, a0, false, wgt, (short)0, acc[0][g][s], false, false);
                acc[1][g][s] = __builtin_amdgcn_wmma_f32_16x16x4_f32(false, a1, false, wgt, (short)0, acc[1][g][s], false, false);
            }
    }
    // ---- child-h part: K = 256..767 ----
    if (HASCH) {  // compile-time: EXEC stays all-ones around WMMA
        const float* __restrict__ ah0 = hbuf + (size_t)(2 * m0 + 1) * HID;  // children rows contiguous
        const float* __restrict__ ah1 = hbuf + (size_t)(2 * m1 + 1) * HID;
        for (int k = 0; k < KH; k += 4) {
            v2f a0 = *(const v2f*)(ah0 + k + ko);
            v2f a1 = *(const v2f*)(ah1 + k + ko);
#pragma unroll
            for (int g = 0; g < NG; ++g)
#pragma unroll
                for (int s = 0; s < 2; ++s) {
                    v2f wgt = *(const v2f*)(bw + (size_t)(256 * g + 16 * s) * KB + KX + k + ko);
                    acc[0][g][s] = __builtin_amdgcn_wmma_f32_16x16x4_f32(false, a0, false, wgt, (short)0, acc[0][g][s], false, false);
                    acc[1][g][s] = __builtin_amdgcn_wmma_f32_16x16x4_f32(false, a1, false, wgt, (short)0, acc[1][g][s], false, false);
                }
        }
    }

    // ---- gates, entirely in registers ----
    // C layout: element j of a v8f = row (j + 8*half) of the tile, column l.
#pragma unroll
    for (int s = 0; s < 2; ++s) {
        const int t  = 32 * wv + 16 * s + l;     // hidden dim owned by this lane
        const float b0 = bias[t];
        const float b1 = bias[HID + t];
        const float b2 = bias[2 * HID + t];
        const float bf = HASCH ? bias[3 * HID + t] : 0.0f;  // f0/f1 share b_f_x[t]
#pragma unroll
        for (int rt = 0; rt < 2; ++rt)
#pragma unroll
            for (int j = 0; j < 8; ++j) {
                const int r = row0 + rt * 16 + j + (half << 3);
                if (r < n) {
                    const int node = levelOff + r;
                    const float gi = sigm(acc[rt][0][s][j] + b0);
                    const float go = sigm(acc[rt][1][s][j] + b1);
                    const float gu = tanhf(acc[rt][2][s][j] + b2);
                    float c;
                    if (HASCH) {
                        const float f0 = sigm(acc[rt][3][s][j] + bf);
                        const float f1 = sigm(acc[rt][4][s][j] + bf);
                        const float cl = cbuf[(size_t)(2 * node + 1) * HID + t];
                        const float cr = cbuf[(size_t)(2 * node + 2) * HID + t];
                        c = gi * gu + f0 * cl + f1 * cr;
                    } else {
                        c = gi * gu;
                    }
                    const float h = go * tanhf(c);
                    cbuf[(size_t)node * HID + t] = c;
                    hbuf[(size_t)node * HID + t] = h;
                }
            }
    }
}

extern "C" void kernel_launch(void* const* d_in, const int* in_sizes, int n_in,
                              void* d_out, int out_size, void* d_ws, size_t ws_size,
                              hipStream_t stream)
{
    const int*   tokens = (const int*)  d_in[0];
    const float* emb    = (const float*)d_in[1];
    const float* Wioux  = (const float*)d_in[2];
    const float* bioux  = (const float*)d_in[3];
    const float* Wfx    = (const float*)d_in[4];
    const float* bfx    = (const float*)d_in[5];
    const float* Wiouh  = (const float*)d_in[6];
    const float* Wfh    = (const float*)d_in[7];

    float* hbuf = (float*)d_out;   // (NNODES x 256) in node-index order == output layout

    // Workspace layout (floats): ~71 MB total
    float* ws   = (float*)d_ws;
    float* Wbig = ws;                                   // 1280*768  (~3.9 MB)
    float* bias = Wbig + (size_t)NCOLS_FULL * KB;       // 1280
    float* cbuf = bias + NCOLS_FULL;                    // NNODES*256 (~67 MB)

    // 1) fuse weights/biases
    {
        int total = NCOLS_FULL * KB;
        prep_weights<<<(total + 255) / 256, 256, 0, stream>>>(
            Wioux, bioux, Wfx, bfx, Wiouh, Wfh, Wbig, bias);
    }

    // 2) leaf level d=15: n=32768, iou only
    {
        int n = 32768, off = 32767;
        fused_level<3, 0><<<(n + 31) / 32, 256, 0, stream>>>(
            tokens, emb, Wbig, bias, hbuf, cbuf, off, n);
    }

    // 3) internal levels d=14..0
    for (int d = 14; d >= 0; --d) {
        int n = 1 << d;
        int off = (1 << d) - 1;
        fused_level<5, 1><<<(n + 31) / 32, 256, 0, stream>>>(
            tokens, emb, Wbig, bias, hbuf, cbuf, off, n);
    }
}